// BlockVectorQuantize_58076547776846
// MI455X (gfx1250) — compile-verified
//
#include <hip/hip_runtime.h>
#include <hip/hip_bf16.h>
#include <math.h>

typedef float v2f __attribute__((ext_vector_type(2)));
typedef float v8f __attribute__((ext_vector_type(8)));

#define NUM_BLOCKS 8
#define DIM 64
#define CODEBOOK 1024
#define BATCH 4
#define SEQ 2048
#define TOKENS (BATCH * SEQ)          /* 8192 */
#define DMODEL (NUM_BLOCKS * DIM)     /* 512  */
#define NTILES (CODEBOOK / 16)        /* 64   */
#define KSTEPS (DIM / 4)              /* 16 wmma f32 16x16x4 steps */
#define MT 2                          /* token sub-tiles (16 rows each) per wave */

// ---------------------------------------------------------------------------
// Kernel 1: per-code squared L2 norms e_sq[k][c]; also zero the commit slots.
// ---------------------------------------------------------------------------
__global__ void vq_esq_init_kernel(const float* __restrict__ cb,
                                   float* __restrict__ esq,
                                   float* __restrict__ commits) {
  const int k = blockIdx.x;        // 0..7
  const int c = threadIdx.x;       // 0..1023
  const float4* row = (const float4*)(cb + ((size_t)k * CODEBOOK + c) * DIM);
  float s = 0.f;
#pragma unroll
  for (int i = 0; i < DIM / 4; ++i) {
    float4 v = row[i];
    s += v.x * v.x + v.y * v.y + v.z * v.z + v.w * v.w;
  }
  esq[k * CODEBOOK + c] = s;
  if (k == 0 && c < NUM_BLOCKS) commits[c] = 0.f;  // d_out is poisoned; init
}

// ---------------------------------------------------------------------------
// Helpers for the WMMA main kernel.
// ---------------------------------------------------------------------------
__device__ __forceinline__ void load_btile(const float* __restrict__ bbase,
                                           int nt, v2f (&b)[KSTEPS]) {
  const float* p = bbase + (size_t)nt * (16 * DIM);
#pragma unroll
  for (int s = 0; s < KSTEPS; ++s) b[s] = *(const v2f*)(p + 4 * s);
}

// One 16-code tile against MT token sub-tiles: 2*MT independent WMMA chains.
__device__ __forceinline__ void vq_tile(const v2f (&a)[MT][KSTEPS],
                                        const v2f (&b)[KSTEPS],
                                        const float* __restrict__ esqk,
                                        int nt, int col,
                                        float (&best_s)[MT][8],
                                        int (&best_i)[MT][8]) {
  const float es = esqk[nt * 16 + col];  // ||code||^2 for this lane's column
  v8f acc[MT][2];
#pragma unroll
  for (int t = 0; t < MT; ++t)
#pragma unroll
    for (int p = 0; p < 2; ++p)
      acc[t][p] = (v8f){0.f, 0.f, 0.f, 0.f, 0.f, 0.f, 0.f, 0.f};

#pragma unroll
  for (int s = 0; s < KSTEPS; s += 2) {
#pragma unroll
    for (int t = 0; t < MT; ++t) {
      acc[t][0] = __builtin_amdgcn_wmma_f32_16x16x4_f32(
          false, a[t][s],     false, b[s],     (short)0, acc[t][0], false, false);
      acc[t][1] = __builtin_amdgcn_wmma_f32_16x16x4_f32(
          false, a[t][s + 1], false, b[s + 1], (short)0, acc[t][1], false, false);
    }
  }
#pragma unroll
  for (int t = 0; t < MT; ++t) {
#pragma unroll
    for (int v = 0; v < 8; ++v) {
      // D layout: VGPR v -> row M = v + half*8, column N = nt*16 + col
      float xe = acc[t][0][v] + acc[t][1][v];
      float score = fmaf(-2.f, xe, es);  // dist minus constant x_sq term
      bool better = score < best_s[t][v];
      best_s[t][v] = better ? score : best_s[t][v];
      best_i[t][v] = better ? nt * 16 + col : best_i[t][v];
    }
  }
}

// ---------------------------------------------------------------------------
// Kernel 2: one wave32 per (32-token tile, block k).
//   xe = A(32x64) * B(64x16codes) via v_wmma_f32_16x16x4_f32; each 4KB B tile
//   feeds 32 WMMAs (M-reuse), double-buffered so loads overlap compute.
// ---------------------------------------------------------------------------
__global__ void __launch_bounds__(32)
vq_main_kernel(const float* __restrict__ x, const float* __restrict__ cb,
               const float* __restrict__ esq, float* __restrict__ out_codes,
               float* __restrict__ out_inds, float* __restrict__ out_commits) {
  const int tile = blockIdx.x;   // 0..255 (32 tokens each)
  const int k    = blockIdx.y;   // 0..7 blocks
  const int lane = threadIdx.x;  // wave32
  const int col  = lane & 15;
  const int half = lane >> 4;
  const int koff = half * 2;     // K = 4s + koff, koff+1 per ISA A/B layout

  const int tok0 = tile * (16 * MT);

  // ---- A tiles: MT x (16 tokens x 64 dims) in WMMA f32 16x16x4 layout ----
  v2f a[MT][KSTEPS];
  float xsq[MT];
#pragma unroll
  for (int t = 0; t < MT; ++t) {
    const float* xrow =
        x + (size_t)(tok0 + t * 16 + col) * DMODEL + (size_t)k * DIM + koff;
    float part = 0.f;
#pragma unroll
    for (int s = 0; s < KSTEPS; ++s) {
      v2f v = *(const v2f*)(xrow + 4 * s);
      a[t][s] = v;
      part += v.x * v.x + v.y * v.y;
    }
    // all lanes now hold x_sq for token row (lane&15) of sub-tile t
    xsq[t] = part + __shfl_xor(part, 16, 32);
  }

  const float* cbk  = cb + (size_t)k * CODEBOOK * DIM;
  const float* esqk = esq + k * CODEBOOK;
  // per-lane B base: column 'col' of each 16-code tile, K pair 'koff'
  const float* bbase = cbk + (size_t)col * DIM + koff;

  float best_s[MT][8];
  int   best_i[MT][8];
#pragma unroll
  for (int t = 0; t < MT; ++t)
#pragma unroll
    for (int v = 0; v < 8; ++v) { best_s[t][v] = INFINITY; best_i[t][v] = 0; }

  // ---- double-buffered ping-pong over 64 code tiles ----
  v2f bcur[KSTEPS], bnext[KSTEPS];
  load_btile(bbase, 0, bcur);
#pragma unroll 1
  for (int nt = 0; nt < NTILES; nt += 2) {
    load_btile(bbase, nt + 1, bnext);                         // fetch tile nt+1
    __builtin_prefetch(bbase + (size_t)(nt + 2) * 16 * DIM, 0, 1);
    vq_tile(a, bcur, esqk, nt, col, best_s, best_i);          // compute tile nt
    load_btile(bbase, (nt + 2 < NTILES) ? nt + 2 : nt, bcur); // fetch nt+2
    vq_tile(a, bnext, esqk, nt + 1, col, best_s, best_i);     // compute nt+1
  }

  // ---- argmin butterfly across the 16 columns in each half-wave ----
#pragma unroll
  for (int t = 0; t < MT; ++t) {
#pragma unroll
    for (int v = 0; v < 8; ++v) {
      float s = best_s[t][v];
      int   i = best_i[t][v];
#pragma unroll
      for (int m = 8; m >= 1; m >>= 1) {
        float s2 = __shfl_xor(s, m, 32);
        int   i2 = __shfl_xor(i, m, 32);
        if (s2 < s || (s2 == s && i2 < i)) { s = s2; i = i2; }
      }
      best_s[t][v] = s;  // min score for row v+half*8 of sub-tile t
      best_i[t][v] = i;  // argmin (lowest index on ties)
    }
  }

  // ---- epilogue: gather winning codes, write indices, commit partial ----
  float commit_acc = 0.f;
#pragma unroll
  for (int t = 0; t < MT; ++t) {
#pragma unroll
    for (int v = 0; v < 8; ++v) {
      const int row   = v + half * 8;
      const int token = tok0 + t * 16 + row;
      const int idx   = best_i[t][v];

      // copy the 64-float code row: 16 lanes x float4 (coalesced)
      float4 q = *(const float4*)(cbk + (size_t)idx * DIM + col * 4);
      *(float4*)(out_codes + (size_t)token * DMODEL + (size_t)k * DIM + col * 4) = q;

      if (col == 0) out_inds[(size_t)token * NUM_BLOCKS + k] = (float)idx;

      // ||x - q||^2 = x_sq + min(e_sq - 2*xe); x_sq of this row via shuffle
      float xs = __shfl(xsq[t], row, 32);
      commit_acc += xs + best_s[t][v];
    }
  }
  // sum the two half-wave partials, single atomic per wave
  float other = __shfl(commit_acc, lane ^ 16, 32);
  if (lane == 0) {
    const float inv_bnd = 1.0f / (float)((size_t)TOKENS * DIM);
    atomicAdd(out_commits + k, (commit_acc + other) * inv_bnd);
  }
}

// ---------------------------------------------------------------------------
extern "C" void kernel_launch(void* const* d_in, const int* in_sizes, int n_in,
                              void* d_out, int out_size, void* d_ws, size_t ws_size,
                              hipStream_t stream) {
  const float* x  = (const float*)d_in[0];  // [4,2048,512] f32
  const float* cb = (const float*)d_in[1];  // [8,1024,64]  f32

  float* out         = (float*)d_out;
  float* out_codes   = out;                                    // 4*2048*512
  float* out_inds    = out + (size_t)TOKENS * DMODEL;          // 4*2048*8
  float* out_commits = out_inds + (size_t)TOKENS * NUM_BLOCKS; // 8

  float* esq = (float*)d_ws;  // 8*1024 floats = 32 KB scratch

  vq_esq_init_kernel<<<dim3(NUM_BLOCKS), dim3(CODEBOOK), 0, stream>>>(
      cb, esq, out_commits);

  dim3 grid(TOKENS / (16 * MT), NUM_BLOCKS);
  vq_main_kernel<<<grid, dim3(32), 0, stream>>>(
      x, cb, esq, out_codes, out_inds, out_commits);
}